// LocalLinear_custom_58780922413890
// MI455X (gfx1250) — compile-verified
//
#include <hip/hip_runtime.h>
#include <stdint.h>

typedef __attribute__((ext_vector_type(2))) float    v2f;
typedef __attribute__((ext_vector_type(8))) float    v8f;
typedef __attribute__((ext_vector_type(4))) uint32_t u32x4;
typedef __attribute__((ext_vector_type(8))) uint32_t u32x8;

#define HWD    32
#define CIN    64
#define COUT   64
#define KS     3
#define WPITCH 72                       // (72*k)%64 = 8k -> B-frag paired rows on disjoint banks
#define APITCH 68                       // via TDM pad: banks 4m+ka -> conflict-free b64 A reads
#define A_ELEMS (64 * APITCH)           // 4352 floats per A buffer, [b][c] rows padded by TDM
#define W_SLAB  (9 * CIN * WPITCH)      // 41472 floats, [tap][c][o], all 9 taps resident
#define LDS_FLOATS (2 * A_ELEMS + W_SLAB)
#define LDS_BYTES  (LDS_FLOATS * 4)     // 200,704 B (< 320 KB/WGP)

// CDNA5 async copy: copies one dword from global memory at base saddr plus offset vaddr to the LDS address in vdst; tracked by ASYNCcnt.
__device__ __forceinline__ void async_b32(uint32_t lds_byte, const void* base, uint32_t byte_off) {
    asm volatile("global_load_async_to_lds_b32 %0, %1, %2"
                 :: "v"(lds_byte), "v"(byte_off), "s"(base) : "memory");
}
__device__ __forceinline__ void wait_async0() {
    asm volatile("s_wait_asynccnt 0x0" ::: "memory");
}

// TDM: one descriptor moves the whole 64x64 x-patch tap (4096 dwords) into LDS,
// inserting 4 pad dwords after every 64 (-> row pitch 68).  D# per ISA 08 §8.3-8.6.
__device__ __forceinline__ void tdm_load_A(const float* xg, uint32_t lds_byte, int yy, int xx)
{
    const uint64_t ga = (uint64_t)(uintptr_t)(xg + yy * HWD + xx);   // tile start VA
    const u32x4 g0 = {
        1u,                                                // count=1, user desc, no gather
        lds_byte,                                          // lds_addr
        (uint32_t)ga,                                      // global_addr[31:0]
        (((uint32_t)(ga >> 32)) & 0x01FFFFFFu) | 0x80000000u  // addr[56:32] | type=2
    };
    const u32x8 g1 = {
        (2u << 16) | (1u << 20) | (5u << 22) | (3u << 25), // dsize=4B, pad_en, every 64dw, +4dw
        (1u << 16),                                        // tensor_dim0 = 1 (lo16)
        (64u << 16),                                       // dim0 hi16=0 | tensor_dim1=64 (lo16)
        (1u << 16),                                        // dim1 hi16=0 | tile_dim0 = 1
        64u | (64u << 16),                                 // tile_dim1=64 (c) | tile_dim2=64 (b)
        (uint32_t)(HWD * HWD),                             // tensor_dim0_stride = 1024 (c)
        0u,                                                // stride0 hi16 | stride1 lo16
        1u                                                 // tensor_dim1_stride = 65536 (b) >>16
    };
    const u32x4 g2 = { 64u, 0u, 0u, 0u };                  // tensor_dim2=64; no dim3
    const u32x4 g3 = { 0u, 0u, 0u, 0u };
    asm volatile("tensor_load_to_lds %0, %1, %2, %3"
                 :: "s"(g0), "s"(g1), "s"(g2), "s"(g3) : "memory");
}

__global__ __launch_bounds__(256)
void locallin_wmma_tdm(const float* __restrict__ xg, const float* __restrict__ wg,
                       const float* __restrict__ bg, float* __restrict__ og)
{
    extern __shared__ float lds[];
    float* Abuf = lds;                        // 2 x A_ELEMS, double buffered
    float* Wall = lds + 2 * A_ELEMS;          // [tap][c][o]
    const uint32_t lds0  = (uint32_t)(uintptr_t)lds;  // low 32 bits of the flat LDS pointer are the LDS offset
    const uint32_t aBase = lds0;
    const uint32_t wBase = lds0 + 2u * A_ELEMS * 4u;

    const int sp = blockIdx.x, y = sp >> 5, x = sp & 31;
    const int t = threadIdx.x, lane = t & 31, wv = t >> 5;
    const int mt   = wv >> 1;                 // batch tile 0..3
    const int half = lane >> 4, l16 = lane & 15;
    const int mrow = mt * 16 + l16;           // batch row for A fragment
    const int n0   = (wv & 1) * 32 + l16;     // two cout tiles per wave (A-frag reuse)
    const int n1   = n0 + 16;

    // ---- stage ALL 9 weight taps once: each (o,c) row = 9 contiguous dwords ----
    {
        const uint32_t wrow = (uint32_t)sp * (KS * KS) * 4u;
        #pragma unroll
        for (int i = 0; i < 16; ++i) {
            const int pair = i * 256 + t;     // 4096 (c,o) pairs / 256 threads
            const int c = pair >> 6;
            const int o = pair & 63;
            const uint32_t src = ((uint32_t)o * (CIN * HWD * HWD * KS * KS)
                                + (uint32_t)c * (HWD * HWD * KS * KS)) * 4u + wrow;
            const uint32_t dst = wBase + ((uint32_t)c * WPITCH + o) * 4u;
            #pragma unroll
            for (int tap = 0; tap < 9; ++tap) // one cacheline burst per (o,c)
                async_b32(dst + (uint32_t)tap * (CIN * WPITCH * 4u), wg, src + tap * 4u);
        }
    }

    // ---- A staging: TDM for interior taps, uniform zero-fill at borders ----
    auto stage_A = [&](int tap, int buf) {
        const int u = tap / 3, v = tap % 3;
        const int yy = y - 1 + u, xx = x - 1 + v;
        const bool ok = ((unsigned)yy < HWD) && ((unsigned)xx < HWD);  // block-uniform
        if (ok) {
            if (t < 32)   // single wave issues the DMA; TENSORcnt is per-wave
                tdm_load_A(xg, aBase + (uint32_t)buf * (A_ELEMS * 4u), yy, xx);
        } else {
            float* A = Abuf + buf * A_ELEMS;
            #pragma unroll
            for (int i = 0; i < 17; ++i) {
                const int idx = i * 256 + t;
                if (idx < A_ELEMS) A[idx] = 0.0f;
            }
        }
    };

    stage_A(0, 0);

    v8f acc0 = {0.f,0.f,0.f,0.f,0.f,0.f,0.f,0.f};
    v8f acc1 = {0.f,0.f,0.f,0.f,0.f,0.f,0.f,0.f};

    int buf = 0;
    for (int tap = 0; tap < 9; ++tap) {
        __builtin_amdgcn_s_wait_tensorcnt(0);  // this tap's TDM tile landed (wave 0)
        wait_async0();                         // weights landed (tap 0; no-op after)
        __syncthreads();                       // everyone sees LDS; old buffer consumed
        if (tap < 8) stage_A(tap + 1, buf ^ 1);// overlap next DMA with compute

        const float* A = Abuf + buf * A_ELEMS;
        const float* W = Wall + tap * (CIN * WPITCH);
        #pragma unroll
        for (int k0 = 0; k0 < CIN; k0 += 4) {
            const int ka = k0 + 2 * half;
            v2f a, b0, b1;
            a.x  = A[mrow * APITCH + ka];  a.y  = A[mrow * APITCH + ka + 1];  // ds_load_b64
            b0.x = W[ka * WPITCH + n0];    b0.y = W[(ka + 1) * WPITCH + n0];
            b1.x = W[ka * WPITCH + n1];    b1.y = W[(ka + 1) * WPITCH + n1];
            acc0 = __builtin_amdgcn_wmma_f32_16x16x4_f32(false, a, false, b0,
                                                         (short)0, acc0, false, false);
            acc1 = __builtin_amdgcn_wmma_f32_16x16x4_f32(false, a, false, b1,
                                                         (short)0, acc1, false, false);
        }
        buf ^= 1;
    }

    // ---- epilogue: D layout -> VGPR r holds M = r (lanes 0-15) / 8+r (lanes 16-31) ----
    const float bias0 = bg[n0 * (HWD * HWD) + sp];
    const float bias1 = bg[n1 * (HWD * HWD) + sp];
    #pragma unroll
    for (int r = 0; r < 8; ++r) {
        const int bb = mt * 16 + half * 8 + r;
        og[(size_t)bb * (COUT * HWD * HWD) + (size_t)n0 * (HWD * HWD) + sp] = acc0[r] + bias0;
        og[(size_t)bb * (COUT * HWD * HWD) + (size_t)n1 * (HWD * HWD) + sp] = acc1[r] + bias1;
    }
}

extern "C" void kernel_launch(void* const* d_in, const int* in_sizes, int n_in,
                              void* d_out, int out_size, void* d_ws, size_t ws_size,
                              hipStream_t stream) {
    const float* x = (const float*)d_in[0];
    const float* w = (const float*)d_in[1];
    const float* b = (const float*)d_in[2];
    float*       o = (float*)d_out;
    (void)in_sizes; (void)n_in; (void)out_size; (void)d_ws; (void)ws_size;

    locallin_wmma_tdm<<<dim3(HWD * HWD), dim3(256), LDS_BYTES, stream>>>(x, w, b, o);
}